// DoReMiLossForProxyTraining_51728586113631
// MI455X (gfx1250) — compile-verified
//
#include <hip/hip_runtime.h>
#include <math.h>

// ---- problem constants (match reference) ----
#define BB 8
#define SS 1024
#define VV 32000
#define ND 8
#define STEP_SIZE 1.0f
#define SMOOTHING 1e-3f

typedef __attribute__((ext_vector_type(4))) float f4;
typedef __attribute__((ext_vector_type(2))) float v2f;
typedef __attribute__((ext_vector_type(8))) float v8f;

// ---------------------------------------------------------------------------
// Kernel 1: one-pass online log-sum-exp per token row (memory-bound, ~45us
// floor at 23.3 TB/s). NT loads: 1GB stream, no reuse, keep L2 clean.
// loss[r] = max + log(sum exp(x-max)) - x[label]
// ---------------------------------------------------------------------------
__global__ __launch_bounds__(256)
void lse_loss_kernel(const float* __restrict__ logits,
                     const int* __restrict__ labels,
                     float* __restrict__ loss_out) {
  const int row = blockIdx.x;
  const int tid = threadIdx.x;
  const f4* __restrict__ p = (const f4*)(logits + (size_t)row * VV);
  const int label = labels[row];

  __shared__ float red[256];
  __shared__ float sLab;

  float m = -__builtin_inff();
  float s = 0.0f;
  float lab = 0.0f;
  bool haveLab = false;

  for (int i = tid; i < VV / 4; i += 256) {
    f4 v = __builtin_nontemporal_load(p + i);   // global_load_b128 th:NT
    int d = label - 4 * i;
    if (d >= 0 && d < 4) {
      lab = (d == 0) ? v.x : (d == 1) ? v.y : (d == 2) ? v.z : v.w;
      haveLab = true;
    }
    // online softmax: rescale running sum only when the max moves
    float lm = fmaxf(fmaxf(v.x, v.y), fmaxf(v.z, v.w));
    if (lm > m) { s *= __expf(m - lm); m = lm; }
    s += __expf(v.x - m) + __expf(v.y - m) + __expf(v.z - m) + __expf(v.w - m);
  }
  if (haveLab) sLab = lab;   // exactly one thread owns the label element

  // block max-reduce
  red[tid] = m;
  __syncthreads();
  for (int off = 128; off > 0; off >>= 1) {
    if (tid < off) red[tid] = fmaxf(red[tid], red[tid + off]);
    __syncthreads();
  }
  const float M = red[0];
  __syncthreads();

  // rescale partial sums to the global max, then sum-reduce
  red[tid] = s * __expf(m - M);
  __syncthreads();
  for (int off = 128; off > 0; off >>= 1) {
    if (tid < off) red[tid] += red[tid + off];
    __syncthreads();
  }
  if (tid == 0) loss_out[row] = M + logf(red[0]) - sLab;
}

// ---------------------------------------------------------------------------
// Kernel 2: per-sample reductions over S=1024 tokens (tiny).
//   ce_num[b] = sum loss*mask, ce_den[b] = sum mask,
//   per_sample[b] = sum max(loss - ref, 0)
// ---------------------------------------------------------------------------
__global__ __launch_bounds__(256)
void per_sample_kernel(const float* __restrict__ loss,
                       const unsigned char* __restrict__ mask,  // jnp bool = 1 byte
                       const float* __restrict__ ref,
                       float* __restrict__ per_sample,
                       float* __restrict__ ce_num,
                       float* __restrict__ ce_den) {
  const int b = blockIdx.x, tid = threadIdx.x;
  __shared__ float r0[256], r1[256], r2[256];
  float ce = 0.f, den = 0.f, ex = 0.f;
  for (int sI = tid; sI < SS; sI += 256) {
    int idx = b * SS + sI;
    float l = loss[idx];
    float mk = mask[idx] ? 1.0f : 0.0f;
    ce += l * mk;
    den += mk;
    float e = l - ref[idx];
    ex += (e > 0.f) ? e : 0.f;
  }
  r0[tid] = ce; r1[tid] = den; r2[tid] = ex;
  __syncthreads();
  for (int off = 128; off > 0; off >>= 1) {
    if (tid < off) {
      r0[tid] += r0[tid + off];
      r1[tid] += r1[tid + off];
      r2[tid] += r2[tid + off];
    }
    __syncthreads();
  }
  if (tid == 0) { ce_num[b] = r0[0]; ce_den[b] = r1[0]; per_sample[b] = r2[0]; }
}

// ---------------------------------------------------------------------------
// Kernel 3: single wave32. segment_sum as an exact one-hot matmul via two
// chained V_WMMA_F32_16X16X4_F32 (K = samples 0..3 then 4..7):
//   D[d][0] = sum_b 1[dom[b]==d] * per_sample[b]   (domain_losses)
//   D[d][1] = sum_b 1[dom[b]==d] * 1               (samples_per_domain)
// Then the 8-wide DoReMi update + all 26 outputs.
// ---------------------------------------------------------------------------
__global__ __launch_bounds__(32)
void doremi_kernel(const int* __restrict__ domain_idxs,
                   const float* __restrict__ domain_weights,
                   const float* __restrict__ per_sample,
                   const float* __restrict__ ce_num,
                   const float* __restrict__ ce_den,
                   float* __restrict__ out) {
  const int lane = threadIdx.x;          // 0..31, EXEC all ones for WMMA
  const int mrow = lane & 15;            // A-matrix row M / B-matrix col N
  const int kb   = (lane >> 4) * 2;      // K pair: lanes 0-15 -> K0,K1; 16-31 -> K2,K3

  // A (16x4, f32): one-hot of domain index, pass1 = samples 0..3, pass2 = 4..7
  int d0 = domain_idxs[kb],     d1 = domain_idxs[kb + 1];
  int d2 = domain_idxs[4 + kb], d3 = domain_idxs[5 + kb];
  v2f a0, a1, b0, b1;
  a0.x = (d0 == mrow) ? 1.f : 0.f;  a0.y = (d1 == mrow) ? 1.f : 0.f;
  a1.x = (d2 == mrow) ? 1.f : 0.f;  a1.y = (d3 == mrow) ? 1.f : 0.f;

  // B (4x16, f32): column 0 = per_sample, column 1 = ones, rest 0
  float p0 = per_sample[kb],     p1 = per_sample[kb + 1];
  float p2 = per_sample[4 + kb], p3 = per_sample[5 + kb];
  b0.x = (mrow == 0) ? p0 : ((mrow == 1) ? 1.f : 0.f);
  b0.y = (mrow == 0) ? p1 : ((mrow == 1) ? 1.f : 0.f);
  b1.x = (mrow == 0) ? p2 : ((mrow == 1) ? 1.f : 0.f);
  b1.y = (mrow == 0) ? p3 : ((mrow == 1) ? 1.f : 0.f);

  v8f c = {};
  c = __builtin_amdgcn_wmma_f32_16x16x4_f32(false, a0, false, b0, (short)0, c, false, false);
  c = __builtin_amdgcn_wmma_f32_16x16x4_f32(false, a1, false, b1, (short)0, c, false, false);

  // C/D 16x16 f32 layout: row d (<8) lives in VGPR d; N = lane (0..15).
  __shared__ float dl[ND], cnt[ND];
  if (lane == 0) { for (int j = 0; j < ND; ++j) dl[j]  = c[j]; }  // column N=0
  if (lane == 1) { for (int j = 0; j < ND; ++j) cnt[j] = c[j]; }  // column N=1
  __syncthreads();

  if (lane == 0) {
    float ceN = 0.f, ceD = 0.f;
    for (int b = 0; b < BB; ++b) { ceN += ce_num[b]; ceD += ce_den[b]; }
    out[0] = ceN / ceD;                                   // ce_loss

    float norm[ND], lw[ND];
    for (int d = 0; d < ND; ++d) {
      float cn = cnt[d];
      norm[d] = (cn > 0.f) ? dl[d] / (cn * (float)SS) : 0.f;  // NaN->0 for empty
      lw[d] = logf(domain_weights[d]) + STEP_SIZE * norm[d];
    }
    float mx = lw[0];
    for (int d = 1; d < ND; ++d) mx = fmaxf(mx, lw[d]);
    float se = 0.f;
    for (int d = 0; d < ND; ++d) se += expf(lw[d] - mx);
    float lse = mx + logf(se);

    float dro = 0.f;
    for (int d = 0; d < ND; ++d) {
      float tw = (1.0f - SMOOTHING) * expf(lw[d] - lse) + SMOOTHING / (float)ND;
      out[2 + d]  = norm[d];          // normalized
      out[10 + d] = tw;               // train_w
      out[18 + d] = cnt[d];           // samples_per_domain (as float value)
      dro += tw * norm[d];
    }
    out[1] = dro;                                         // dro_loss
  }
}

// ---------------------------------------------------------------------------
extern "C" void kernel_launch(void* const* d_in, const int* in_sizes, int n_in,
                              void* d_out, int out_size, void* d_ws, size_t ws_size,
                              hipStream_t stream) {
  const float*         logits = (const float*)d_in[0];         // [B,S,V] f32
  const int*           labels = (const int*)d_in[1];           // [B,S] i32
  const unsigned char* mask   = (const unsigned char*)d_in[2]; // [B,S] bool
  const int*           dom    = (const int*)d_in[3];           // [B] i32
  const float*         ref    = (const float*)d_in[4];         // [B,S] f32
  const float*         dw     = (const float*)d_in[5];         // [ND] f32

  float* wsf        = (float*)d_ws;
  float* loss       = wsf;                 // B*S = 8192 floats
  float* per_sample = wsf + BB * SS;       // 8
  float* ce_num     = per_sample + BB;     // 8
  float* ce_den     = ce_num + BB;         // 8
  float* out        = (float*)d_out;       // 26 floats

  lse_loss_kernel<<<BB * SS, 256, 0, stream>>>(logits, labels, loss);
  per_sample_kernel<<<BB, 256, 0, stream>>>(loss, mask, ref, per_sample, ce_num, ce_den);
  doremi_kernel<<<1, 32, 0, stream>>>(dom, dw, per_sample, ce_num, ce_den, out);
}